// GE2ELoss_75625784147978
// MI455X (gfx1250) — compile-verified
//
#include <hip/hip_runtime.h>
#include <math.h>

typedef __attribute__((ext_vector_type(2))) float v2f;
typedef __attribute__((ext_vector_type(8))) float v8f;

#define S_   512
#define U_   32
#define D_   512
#define EPSF 1e-6f

// ---------------------------------------------------------------------------
// Kernel 0: zero the scalar output (we accumulate with atomicAdd)
// ---------------------------------------------------------------------------
__global__ void ge2e_zero_kernel(float* out) { out[0] = 0.0f; }

// ---------------------------------------------------------------------------
// Kernel 1: centroids, stored transposed centT[d*S + s], plus cn[s] = |centroid|
// ---------------------------------------------------------------------------
__global__ __launch_bounds__(256)
void ge2e_centroids_kernel(const float* __restrict__ dvecs,
                           float* __restrict__ centT,
                           float* __restrict__ cn) {
  __shared__ float red[256];
  const int s = blockIdx.x;
  const int t = threadIdx.x;
  float sq = 0.0f;
#pragma unroll
  for (int i = 0; i < 2; ++i) {
    const int d = t + i * 256;
    const float* p = dvecs + (size_t)s * U_ * D_ + d;
    float sum = 0.0f;
#pragma unroll
    for (int u = 0; u < U_; ++u) sum += p[u * D_];
    const float c = sum * (1.0f / (float)U_);
    centT[d * S_ + s] = c;
    sq += c * c;
  }
  red[t] = sq;
  __syncthreads();
  for (int off = 128; off > 0; off >>= 1) {
    if (t < off) red[t] += red[t + off];
    __syncthreads();
  }
  if (t == 0) cn[s] = sqrtf(red[0]);
}

// ---------------------------------------------------------------------------
// Kernel 2: per (s,u) row stats -> dn[m] = |d|, ss[m] = leave-one-out cosine.
//   loo = (U·c − d)/(U−1);  dot(d,loo) = (U·dc − dd)/(U−1)
//   |loo|² = (U²·cc − 2U·dc + dd)/(U−1)²
// ---------------------------------------------------------------------------
__global__ __launch_bounds__(256)
void ge2e_rowstats_kernel(const float* __restrict__ dvecs,
                          const float* __restrict__ centT,
                          float* __restrict__ dn,
                          float* __restrict__ ss) {
  __shared__ float r0[256], r1[256], r2[256];
  const int m = blockIdx.x;
  const int s = m >> 5;            // m / U_
  const int t = threadIdx.x;
  float dd = 0.0f, dc = 0.0f, cc = 0.0f;
#pragma unroll
  for (int i = 0; i < 2; ++i) {
    const int d = t + i * 256;
    const float dv = dvecs[(size_t)m * D_ + d];
    const float c  = centT[d * S_ + s];
    dd += dv * dv;
    dc += dv * c;
    cc += c * c;
  }
  r0[t] = dd; r1[t] = dc; r2[t] = cc;
  __syncthreads();
  for (int off = 128; off > 0; off >>= 1) {
    if (t < off) { r0[t] += r0[t + off]; r1[t] += r1[t + off]; r2[t] += r2[t + off]; }
    __syncthreads();
  }
  if (t == 0) {
    dd = r0[0]; dc = r1[0]; cc = r2[0];
    const float inv_um1 = 1.0f / (float)(U_ - 1);
    const float dot_loo = ((float)U_ * dc - dd) * inv_um1;
    const float ln2 = ((float)U_ * (float)U_ * cc - 2.0f * (float)U_ * dc + dd) * inv_um1 * inv_um1;
    const float dnv = sqrtf(dd);
    const float lnv = sqrtf(fmaxf(ln2, 0.0f));
    dn[m] = dnv;
    ss[m] = dot_loo / fmaxf(dnv * lnv, EPSF);
  }
}

// ---------------------------------------------------------------------------
// Kernel 3: fused GEMM (WMMA f32 16x16x4) + cosine scale + diagonal patch +
// log-softmax + loss accumulation.
//
// Block = 256 threads = 8 waves, owns rows [m0, m0+32) (two 16-row WMMA
// stripes, so each B fragment is reused by 2 WMMAs -> half the L2 B-traffic).
// Wave w owns columns [w*64, w*64+64) as 4 N-tiles. K-loop: D=512 step 4.
//
// A frag (ISA 16x4 f32): lane = 16*half + l16 holds A(m+l16, k0+2*half+{0,1}).
// B frag symmetric with N in place of M, read from centT[k*S + n].
// C/D: element r of lane = C(m0 + 16*sub + r + 8*half, n0 + l16).
// ---------------------------------------------------------------------------
__global__ __launch_bounds__(256)
void ge2e_wmma_kernel(const float* __restrict__ dvecs,
                      const float* __restrict__ centT,
                      const float* __restrict__ cn,
                      const float* __restrict__ dn,
                      const float* __restrict__ ss,
                      const float* __restrict__ wptr,
                      const float* __restrict__ bptr,
                      float* __restrict__ out) {
  __shared__ float lds[32 * 513];   // logits stripe, padded rows (~66 KB)

  const int lane = threadIdx.x & 31;
  const int wave = threadIdx.x >> 5;
  const int half = lane >> 4;       // 0 or 1
  const int l16  = lane & 15;
  const int m0   = blockIdx.x * 32;
  const float w = wptr[0];
  const float b = bptr[0];

  v8f acc[2][4];
#pragma unroll
  for (int sub = 0; sub < 2; ++sub)
#pragma unroll
    for (int t = 0; t < 4; ++t) acc[sub][t] = (v8f)(0.0f);

  // A: dvecs viewed as [16384 x 512] row-major; two 16-row stripes
  const float* Arow0 = dvecs + (size_t)(m0 + l16) * D_ + 2 * half;
  const float* Arow1 = Arow0 + (size_t)16 * D_;
  const int nbase = wave * 64 + l16;
  const float* Bc = centT + nbase;

  for (int k0 = 0; k0 < D_; k0 += 4) {
    v2f a0, a1;
    a0.x = Arow0[k0];
    a0.y = Arow0[k0 + 1];
    a1.x = Arow1[k0];
    a1.y = Arow1[k0 + 1];
    // speculative prefetch of the A streams ~1KB ahead (dropped if OOB)
    __builtin_prefetch(Arow0 + k0 + 256, 0, 0);
    __builtin_prefetch(Arow1 + k0 + 256, 0, 0);
    const int kk = (k0 + 2 * half) * S_;
#pragma unroll
    for (int t = 0; t < 4; ++t) {
      v2f bf;
      bf.x = Bc[kk + t * 16];
      bf.y = Bc[kk + S_ + t * 16];
      acc[0][t] = __builtin_amdgcn_wmma_f32_16x16x4_f32(
          false, a0, false, bf, (short)0, acc[0][t], false, false);
      acc[1][t] = __builtin_amdgcn_wmma_f32_16x16x4_f32(
          false, a1, false, bf, (short)0, acc[1][t], false, false);
    }
  }

  // Hoist row norms into registers: rows rbase..rbase+7 of each 16-row stripe
  const int rbase = 8 * half;
  float dnr[2][8];
#pragma unroll
  for (int sub = 0; sub < 2; ++sub)
#pragma unroll
    for (int r = 0; r < 8; ++r) dnr[sub][r] = dn[m0 + 16 * sub + rbase + r];

  // Cosine scale (fast rcp), diagonal patch, w/b, stash logits in LDS
#pragma unroll
  for (int t = 0; t < 4; ++t) {
    const int n = nbase + t * 16;       // global column (speaker j)
    const float cnv = cn[n];
#pragma unroll
    for (int sub = 0; sub < 2; ++sub) {
#pragma unroll
      for (int r = 0; r < 8; ++r) {
        const int rloc = 16 * sub + rbase + r;   // 0..31 local row
        const int m    = m0 + rloc;              // global row
        const int srow = m >> 5;                 // speaker of this row
        const float inv = __builtin_amdgcn_rcpf(fmaxf(dnr[sub][r] * cnv, EPSF));
        float sim = acc[sub][t][r] * inv;
        if (n == srow) sim = ss[m];              // leave-one-out diagonal
        lds[rloc * 513 + n] = sim * w + b;
      }
    }
  }
  __syncthreads();

  // Per-wave row log-softmax: wave handles rows 4*wave .. 4*wave+3
#pragma unroll
  for (int rr = 0; rr < 4; ++rr) {
    const int rloc = wave * 4 + rr;
    const int m    = m0 + rloc;
    const int srow = m >> 5;
    float mx = -INFINITY;
    for (int n = lane; n < S_; n += 32) mx = fmaxf(mx, lds[rloc * 513 + n]);
#pragma unroll
    for (int off = 16; off > 0; off >>= 1) mx = fmaxf(mx, __shfl_xor(mx, off, 32));
    float se = 0.0f;
    for (int n = lane; n < S_; n += 32) se += __expf(lds[rloc * 513 + n] - mx);
#pragma unroll
    for (int off = 16; off > 0; off >>= 1) se += __shfl_xor(se, off, 32);
    if (lane == 0) {
      const float lse = mx + __logf(se);
      atomicAdd(out, lse - lds[rloc * 513 + srow]);   // -(diag - lse)
    }
  }
}

// ---------------------------------------------------------------------------
// Launch
// ---------------------------------------------------------------------------
extern "C" void kernel_launch(void* const* d_in, const int* in_sizes, int n_in,
                              void* d_out, int out_size, void* d_ws, size_t ws_size,
                              hipStream_t stream) {
  (void)in_sizes; (void)n_in; (void)out_size; (void)ws_size;
  const float* dvecs = (const float*)d_in[0];
  const float* w     = (const float*)d_in[1];
  const float* b     = (const float*)d_in[2];
  float* out = (float*)d_out;

  // Workspace layout (floats): centT[D*S] | cn[S] | dn[S*U] | ss[S*U]
  float* centT = (float*)d_ws;
  float* cn    = centT + (size_t)D_ * S_;
  float* dn    = cn + S_;
  float* ss    = dn + (size_t)S_ * U_;

  ge2e_zero_kernel<<<1, 1, 0, stream>>>(out);
  ge2e_centroids_kernel<<<S_, 256, 0, stream>>>(dvecs, centT, cn);
  ge2e_rowstats_kernel<<<S_ * U_, 256, 0, stream>>>(dvecs, centT, dn, ss);
  ge2e_wmma_kernel<<<(S_ * U_) / 32, 256, 0, stream>>>(dvecs, centT, cn, dn, ss, w, b, out);
}